// Coo2FulSimple_231928234116
// MI455X (gfx1250) — compile-verified
//
#include <hip/hip_runtime.h>

typedef float v2f __attribute__((ext_vector_type(2)));
typedef float v8f __attribute__((ext_vector_type(8)));

#define RC2  36.0f     // rc^2, rc = 6.0
#define PEN  1e30f     // entity penalty folded into norms

// ---------------------------------------------------------------------------
// Phase 0: pos_xyz[b,n] = pos[b,n,:] @ cel[b]  (w = entity penalty)
//          sft_xyz[b,s] = sft_cel[s,:] @ cel[b]
// ---------------------------------------------------------------------------
__global__ void prep_kernel(const float* __restrict__ pos,
                            const float* __restrict__ cel,
                            const unsigned char* __restrict__ ent,
                            const int* __restrict__ sft,
                            float4* __restrict__ posx,   // [B*N]
                            float4* __restrict__ sftx,   // [B*S]
                            int B, int N, int S) {
  int t = blockIdx.x * blockDim.x + threadIdx.x;
  int total = B * N + B * S;
  if (t >= total) return;
  if (t < B * N) {
    int b = t / N;
    const float* c = cel + b * 9;
    const float* p = pos + (size_t)t * 3;
    float x = p[0] * c[0] + p[1] * c[3] + p[2] * c[6];
    float y = p[0] * c[1] + p[1] * c[4] + p[2] * c[7];
    float z = p[0] * c[2] + p[1] * c[5] + p[2] * c[8];
    float w = ent[t] ? 0.0f : PEN;
    posx[t] = make_float4(x, y, z, w);
  } else {
    int k = t - B * N;
    int b = k / S;
    int s = k - b * S;
    const float* c = cel + b * 9;
    float s0 = (float)sft[s * 3 + 0];
    float s1 = (float)sft[s * 3 + 1];
    float s2 = (float)sft[s * 3 + 2];
    float x = s0 * c[0] + s1 * c[3] + s2 * c[6];
    float y = s0 * c[1] + s1 * c[4] + s2 * c[7];
    float z = s0 * c[2] + s1 * c[5] + s2 * c[8];
    sftx[k] = make_float4(x, y, z, 0.0f);
  }
}

// ---------------------------------------------------------------------------
// Phase 1: one wave32 per 16x16 pair tile.
//   sod tile = V_WMMA_F32_16X16X4_F32( A = ri (16x4, K<=3),
//                                      B = -2*rj' (4x16),
//                                      C = |ri|^2 + |rj'|^2 )
// CN = compile-time N (0 -> runtime fallback). With CN fixed, all per-row
// store offsets fold into the global-store 24-bit immediate offset field.
// ---------------------------------------------------------------------------
template <int CN>
__global__ void __launch_bounds__(256) pair_kernel(
    const float4* __restrict__ posx,   // [B*N] xyz + ent penalty in w
    const float4* __restrict__ sftx,   // [B*S]
    const int*    __restrict__ sft,    // [S*3]
    float* __restrict__ vec_out,       // [B,S,N,N,3]
    float* __restrict__ sod_out,       // [B,S,N,N]
    float* __restrict__ mask_out,      // [B,S,N,N] as 0/1 float
    int B, int Nrt, int S, int NT, int TJG) {
  const int N = CN ? CN : Nrt;

  __shared__ float4 sRi[16];           // ri tile shared by all 8 waves (same ti)

  const int wave = threadIdx.x >> 5;   // 0..7
  const int lane = threadIdx.x & 31;
  const int col  = lane & 15;          // N (column) index within tile
  const int half = lane >> 4;          // 0: lanes 0-15, 1: lanes 16-31

  int blockId = blockIdx.x;
  int tjg = blockId % TJG;
  int ti  = (blockId / TJG) % NT;
  int bs  =  blockId / (TJG * NT);
  int s   = bs % S;
  int b   = bs / S;
  const int i0 = ti * 16;

  // stage the shared 16-row ri tile through LDS
  if (threadIdx.x < 16) {
    sRi[threadIdx.x] = posx[(size_t)b * N + i0 + threadIdx.x];
  }
  __syncthreads();

  int tj = tjg * 8 + wave;             // wave-uniform
  if (tj >= NT) return;
  const int j0 = tj * 16;
  const int j  = j0 + col;

  // shift vector + zero-shift flag
  float4 sh = sftx[b * S + s];
  bool zs = (sft[s * 3] == 0) && (sft[s * 3 + 1] == 0) && (sft[s * 3 + 2] == 0);

  // column atom j' = rj + shift; norm carries the entity penalty
  float4 pj = posx[(size_t)b * N + j];
  float xj = pj.x + sh.x;
  float yj = pj.y + sh.y;
  float zj = pj.z + sh.z;
  float nrm_j = fmaf(xj, xj, fmaf(yj, yj, fmaf(zj, zj, pj.w)));

  // pull this lane's 8 ri rows (M = r + 8*half) into registers once
  float4 rows[8];
#pragma unroll
  for (int r = 0; r < 8; ++r) rows[r] = sRi[r + 8 * half];

  // A-matrix 16x4 f32: lanes 0-15 -> K=0,1 ; lanes 16-31 -> K=2,3 (row m=col)
  float4 rim = sRi[col];
  v2f a;
  a[0] = half ? rim.z : rim.x;
  a[1] = half ? 0.0f  : rim.y;

  // B-matrix 4x16 (transposed mirror): column n=col, rows K per half
  v2f bm;
  bm[0] = half ? (-2.0f * zj) : (-2.0f * xj);
  bm[1] = half ? 0.0f         : (-2.0f * yj);

  // C accumulator: C[m][n] = |ri(m)|^2 + |rj'(n)|^2
  v8f c;
#pragma unroll
  for (int r = 0; r < 8; ++r) {
    float4 rm = rows[r];
    c[r] = fmaf(rm.x, rm.x, fmaf(rm.y, rm.y, fmaf(rm.z, rm.z, rm.w + nrm_j)));
  }

  // D = A x B + C  ->  full 16x16 sod tile in one WMMA
  v8f d = __builtin_amdgcn_wmma_f32_16x16x4_f32(
      /*neg_a=*/false, a, /*neg_b=*/false, bm,
      /*c_mod=*/(short)0, c, /*reuse_a=*/false, /*reuse_b=*/false);

  // hoisted base pointers; per-row offsets are compile-time constants (CN path)
  const size_t NN = (size_t)N * N;
  const size_t rowBase = ((size_t)b * S + s) * NN + (size_t)(i0 + 8 * half) * N + j;
  float* __restrict__ sodp = sod_out  + rowBase;
  float* __restrict__ mskp = mask_out + rowBase;
  float* __restrict__ vecp = vec_out  + rowBase * 3;

  const int jj = j - i0 - 8 * half;    // row r is a self pair iff r == jj

#pragma unroll
  for (int r = 0; r < 8; ++r) {
    float4 rm = rows[r];
    float vx = rm.x - xj;
    float vy = rm.y - yj;
    float vz = rm.z - zj;
    float sod = d[r];
    // branchless 0/1 mask: inside cutoff (entity penalty pushes sod >> rc^2)
    // and not a self pair at the zero shift
    float msel = ((sod < RC2) & !(zs & (r == jj))) ? 1.0f : 0.0f;
    __builtin_nontemporal_store(sod * msel, sodp + (size_t)r * N);
    __builtin_nontemporal_store(msel,       mskp + (size_t)r * N);
    float3 v3 = make_float3(vx * msel, vy * msel, vz * msel);
    *(float3*)(vecp + (size_t)r * 3 * N) = v3;   // global_store_b96, coalesced
  }
}

// ---------------------------------------------------------------------------
extern "C" void kernel_launch(void* const* d_in, const int* in_sizes, int n_in,
                              void* d_out, int out_size, void* d_ws, size_t ws_size,
                              hipStream_t stream) {
  const float*         pos = (const float*)d_in[0];          // [B,N,3]
  const float*         cel = (const float*)d_in[1];          // [B,3,3]
  const unsigned char* ent = (const unsigned char*)d_in[2];  // [B,N] bool
  const int*           sft = (const int*)d_in[3];            // [S,3]

  const int B = in_sizes[1] / 9;
  const int N = in_sizes[0] / (3 * B);
  const int S = in_sizes[3] / 3;
  const int NT = N / 16;               // N = 768 -> 48 tiles
  const int TJG = (NT + 7) / 8;        // 8 waves (tiles) per 256-thread block

  // workspace: posx [B*N] float4, then sftx [B*S] float4
  float4* posx = (float4*)d_ws;
  float4* sftx = posx + (size_t)B * N;

  // output: vec | sod | mask concatenated flat
  float* vec_out  = (float*)d_out;
  float* sod_out  = vec_out + (size_t)B * S * N * N * 3;
  float* mask_out = sod_out + (size_t)B * S * N * N;

  int prepTotal = B * N + B * S;
  prep_kernel<<<(prepTotal + 255) / 256, 256, 0, stream>>>(
      pos, cel, ent, sft, posx, sftx, B, N, S);

  int grid = B * S * NT * TJG;         // 2*27*48*6 = 15552 blocks
  if (N == 768) {
    pair_kernel<768><<<grid, 256, 0, stream>>>(
        posx, sftx, sft, vec_out, sod_out, mask_out, B, N, S, NT, TJG);
  } else {
    pair_kernel<0><<<grid, 256, 0, stream>>>(
        posx, sftx, sft, vec_out, sod_out, mask_out, B, N, S, NT, TJG);
  }
}